// Detect_87917980549461
// MI455X (gfx1250) — compile-verified
//
#include <hip/hip_runtime.h>
#include <cstdint>

// Problem constants (match reference)
#define B_  16
#define A_  16384
#define C_  81
#define K_  50
#define THREADS 256
#define PER (A_ / THREADS)      // 64 anchors per thread
#define CONF_TH 0.5f
#define IOU_TH  0.5f

// Low 32 bits of a generic pointer to LDS == wave-relative LDS byte offset
// (flat aperture mapping keeps the LDS offset in the low 32 bits, CDNA5 ISA).
__device__ __forceinline__ uint32_t lds_off_u32(const void* p) {
    return (uint32_t)(uintptr_t)p;
}

// ---------------------------------------------------------------------------
// Kernel 1: SSD decode + clip.  boxes[b,a] = clip(decode(loc[b,a], prior[a]))
// ---------------------------------------------------------------------------
__global__ __launch_bounds__(256)
void decode_kernel(const float* __restrict__ loc,      // [B,A,4]
                   const float* __restrict__ anchors,  // [A,4]
                   float4* __restrict__ boxes)         // [B*A]
{
    int i = blockIdx.x * blockDim.x + threadIdx.x;
    if (i >= B_ * A_) return;
    int a = i & (A_ - 1);

    float4 pr = ((const float4*)anchors)[a];
    float4 ld = ((const float4*)loc)[i];

    float cx = pr.x + ld.x * 0.1f * pr.z;
    float cy = pr.y + ld.y * 0.1f * pr.w;
    float w  = pr.z * expf(ld.z * 0.2f);
    float h  = pr.w * expf(ld.w * 0.2f);

    float x1 = cx - 0.5f * w;
    float y1 = cy - 0.5f * h;
    float x2 = x1 + w;          // reference computes x2y2 = x1y1 + wh
    float y2 = y1 + h;

    x1 = fminf(fmaxf(x1, 0.f), 1.f);
    y1 = fminf(fmaxf(y1, 0.f), 1.f);
    x2 = fminf(fmaxf(x2, 0.f), 1.f);
    y2 = fminf(fmaxf(y2, 0.f), 1.f);

    boxes[i] = make_float4(x1, y1, x2, y2);
}

// ---------------------------------------------------------------------------
// Kernel 2: one workgroup per (b,c).  Scores live in LDS (async-gathered),
// 50 sequential argmax+suppress steps, thread 0 emits [K,5] rows.
// ---------------------------------------------------------------------------
__global__ __launch_bounds__(THREADS)
void nms_kernel(const float*  __restrict__ conf,   // [B,A,C]
                const float4* __restrict__ boxes,  // [B*A]
                float* __restrict__ out)           // [B,C,K,5]
{
    extern __shared__ float smem[];
    float*    s      = smem;                         // A_ masked scores
    float*    rv     = smem + A_;                    // 8 per-wave best values
    unsigned* ri     = (unsigned*)(smem + A_ + 8);   // 8 per-wave best indices
    float*    bestv  = smem + A_ + 16;               // combined best value
    unsigned* besti  = (unsigned*)(smem + A_ + 17);  // combined best index

    const int tid = threadIdx.x;
    const int bc  = blockIdx.x;
    const int b   = bc / C_;
    const int c   = bc % C_;
    const float NEGINF = -__builtin_inff();

    // --- async strided gather: conf[b, :, c] -> LDS (CDNA5 ASYNCcnt path) ---
    const float* gbase = conf + ((size_t)b * A_) * C_ + c;
#pragma unroll 8
    for (int j = 0; j < PER; ++j) {
        int a = j * THREADS + tid;                    // coalesced LDS layout
        const float* gp = gbase + (size_t)a * C_;     // stride 81 floats
        uint32_t lo = lds_off_u32(&s[a]);
        asm volatile("global_load_async_to_lds_b32 %0, %1, off"
                     :: "v"(lo), "v"(gp) : "memory");
    }
    asm volatile("s_wait_asynccnt 0" ::: "memory");   // drain this wave's DMA
    __syncthreads();                                  // publish to all 8 waves

    // --- confidence threshold mask (valid-box mask is all-true: min_size=0) ---
#pragma unroll 8
    for (int j = 0; j < PER; ++j) {
        int a = j * THREADS + tid;
        float v = s[a];
        s[a] = (v > CONF_TH) ? v : NEGINF;
    }
    __syncthreads();

    const float4* bx   = boxes + (size_t)b * A_;
    float*        outp = out + (size_t)bc * K_ * 5;

    for (int k = 0; k < K_; ++k) {
        // ---- per-thread argmax over its strided slice (first-index wins) ----
        float    bvv = NEGINF;
        unsigned bii = 0xFFFFFFFFu;
#pragma unroll 8
        for (int j = 0; j < PER; ++j) {
            int a = j * THREADS + tid;
            float v = s[a];
            if (v > bvv || (v == bvv && (unsigned)a < bii)) { bvv = v; bii = (unsigned)a; }
        }
        // ---- wave32 butterfly reduce (lower index breaks ties, jnp.argmax) ----
#pragma unroll
        for (int m = 16; m >= 1; m >>= 1) {
            float    ov = __shfl_xor(bvv, m, 32);
            unsigned oi = __shfl_xor(bii, m, 32);
            if (ov > bvv || (ov == bvv && oi < bii)) { bvv = ov; bii = oi; }
        }
        int wave = tid >> 5;
        if ((tid & 31) == 0) { rv[wave] = bvv; ri[wave] = bii; }
        __syncthreads();

        // ---- cross-wave combine + output row + kill selected score ----
        if (tid == 0) {
            float    v  = rv[0];
            unsigned i0 = ri[0];
#pragma unroll
            for (int w = 1; w < THREADS / 32; ++w) {
                float ov = rv[w]; unsigned oi = ri[w];
                if (ov > v || (ov == v && oi < i0)) { v = ov; i0 = oi; }
            }
            *bestv = v;
            *besti = i0;
            s[i0]  = NEGINF;                          // s.at[i].set(-inf)

            bool   take = v > NEGINF;
            float4 bb   = take ? bx[i0] : make_float4(0.f, 0.f, 0.f, 0.f);
            outp[k * 5 + 0] = take ? v : 0.f;         // conf value == masked value
            outp[k * 5 + 1] = bb.x;
            outp[k * 5 + 2] = bb.y;
            outp[k * 5 + 3] = bb.z;
            outp[k * 5 + 4] = bb.w;
        }
        __syncthreads();

        // ---- suppression: iou(best, a) > 0.5 -> -inf (skipped when !take) ----
        float    v  = *bestv;
        unsigned i0 = *besti;
        if (v > NEGINF) {
            float4 bb = bx[i0];                       // uniform address: 1 fetch
            float areaB = (bb.z - bb.x) * (bb.w - bb.y);
#pragma unroll 4
            for (int j = 0; j < PER; ++j) {
                int a = j * THREADS + tid;
                float sv = s[a];
                if (sv > NEGINF) {                    // dead lanes: no L2 traffic
                    float4 q = bx[a];
                    float xx1 = fmaxf(bb.x, q.x);
                    float yy1 = fmaxf(bb.y, q.y);
                    float xx2 = fminf(bb.z, q.z);
                    float yy2 = fminf(bb.w, q.w);
                    float inter = fmaxf(xx2 - xx1, 0.f) * fmaxf(yy2 - yy1, 0.f);
                    float areaA = (q.z - q.x) * (q.w - q.y);
                    float iou   = inter / (areaB + areaA - inter);
                    if (iou > IOU_TH) s[a] = NEGINF;
                }
            }
        }
        __syncthreads();
    }
}

// ---------------------------------------------------------------------------
extern "C" void kernel_launch(void* const* d_in, const int* in_sizes, int n_in,
                              void* d_out, int out_size, void* d_ws, size_t ws_size,
                              hipStream_t stream) {
    const float* conf = (const float*)d_in[0];   // [B,A,C]
    const float* loc  = (const float*)d_in[1];   // [B,A,4]
    const float* anch = (const float*)d_in[2];   // [A,4]
    float4* boxes = (float4*)d_ws;               // 4 MB scratch: decoded boxes

    decode_kernel<<<(B_ * A_ + 255) / 256, 256, 0, stream>>>(loc, anch, boxes);

    size_t shmem = (size_t)(A_ + 32) * sizeof(float);   // 64 KB scores + scratch
    nms_kernel<<<B_ * C_, THREADS, shmem, stream>>>(conf, boxes, (float*)d_out);
}